// RoPEAttention_2594160247014
// MI455X (gfx1250) — compile-verified
//
#include <hip/hip_runtime.h>

typedef float v2f __attribute__((ext_vector_type(2)));
typedef float v8f __attribute__((ext_vector_type(8)));

// D = A(16x4 f32) * B(4x16 f32) + C(16x16 f32), wave32
#define WMMA_F32(a, b, c) \
    __builtin_amdgcn_wmma_f32_16x16x4_f32(false, (a), false, (b), (short)0, (c), false, false)

constexpr int Bsz    = 4;
constexpr int Nseq   = 2048;
constexpr int Cdim   = 1024;
constexpr int Hn     = 16;
constexpr int HD     = 64;            // head dim
constexpr int ThreeC = 3 * Cdim;      // 3072
constexpr int Mrows  = Bsz * Nseq;    // 8192

// ---------------------------------------------------------------------------
// GEMM: Y[m][n] = sum_k X[m][k] * W[n][k]  (+ bias[n])
// One wave -> 32(M) x 64(N) tile: 2 A-frags x 4 B-frags, 8 WMMA per K-step.
// Software-pipelined: fragments for k0+4 are loaded before the k0 WMMAs issue,
// so waits become s_wait_loadcnt <=6 instead of 0 (latency hidden).
// A-frag (16x4 f32): lanes 0-15 hold row m, K=k0..k0+1; lanes 16-31 K=k0+2..k0+3
// B-frag (4x16 f32): lanes 0-15 hold col n, K=k0..k0+1; lanes 16-31 K=k0+2..k0+3
// C (16x16 f32): vgpr r, lane l -> M = r + 8*(l>=16), N = l&15
// ---------------------------------------------------------------------------
__global__ __launch_bounds__(256)
void gemm_xwt_kernel(const float* __restrict__ X, const float* __restrict__ W,
                     const float* __restrict__ bias, float* __restrict__ Y,
                     int M, int Nd, int K)
{
    const int lane = threadIdx.x & 31;
    const int wid  = blockIdx.x * (blockDim.x >> 5) + (threadIdx.x >> 5);
    const int tilesN = Nd >> 6;
    const int m0 = (wid / tilesN) * 32;
    const int n0 = (wid % tilesN) * 64;
    const int hlf = lane >> 4;        // 0 or 1 (which lane half)
    const int l16 = lane & 15;

    v8f zero = {};
    v8f acc[2][4];
#pragma unroll
    for (int i = 0; i < 2; i++)
#pragma unroll
        for (int j = 0; j < 4; j++) acc[i][j] = zero;

    const float* arow0 = X + (size_t)(m0 + l16) * K;
    const float* arow1 = X + (size_t)(m0 + 16 + l16) * K;
    const float* brow0 = W + (size_t)(n0 +  0 + l16) * K;
    const float* brow1 = W + (size_t)(n0 + 16 + l16) * K;
    const float* brow2 = W + (size_t)(n0 + 32 + l16) * K;
    const float* brow3 = W + (size_t)(n0 + 48 + l16) * K;

    const int ko0 = 2 * hlf;
    v2f a0 = *(const v2f*)(arow0 + ko0);
    v2f a1 = *(const v2f*)(arow1 + ko0);
    v2f b0 = *(const v2f*)(brow0 + ko0);
    v2f b1 = *(const v2f*)(brow1 + ko0);
    v2f b2 = *(const v2f*)(brow2 + ko0);
    v2f b3 = *(const v2f*)(brow3 + ko0);

    for (int k0 = 4; k0 < K; k0 += 4) {
        const int kn = k0 + ko0;
        // prefetch next iteration's fragments (overlaps with WMMA chain below)
        v2f na0 = *(const v2f*)(arow0 + kn);
        v2f na1 = *(const v2f*)(arow1 + kn);
        v2f nb0 = *(const v2f*)(brow0 + kn);
        v2f nb1 = *(const v2f*)(brow1 + kn);
        v2f nb2 = *(const v2f*)(brow2 + kn);
        v2f nb3 = *(const v2f*)(brow3 + kn);

        acc[0][0] = WMMA_F32(a0, b0, acc[0][0]);
        acc[1][0] = WMMA_F32(a1, b0, acc[1][0]);
        acc[0][1] = WMMA_F32(a0, b1, acc[0][1]);
        acc[1][1] = WMMA_F32(a1, b1, acc[1][1]);
        acc[0][2] = WMMA_F32(a0, b2, acc[0][2]);
        acc[1][2] = WMMA_F32(a1, b2, acc[1][2]);
        acc[0][3] = WMMA_F32(a0, b3, acc[0][3]);
        acc[1][3] = WMMA_F32(a1, b3, acc[1][3]);

        a0 = na0; a1 = na1; b0 = nb0; b1 = nb1; b2 = nb2; b3 = nb3;
    }
    // epilogue K-step
    acc[0][0] = WMMA_F32(a0, b0, acc[0][0]);
    acc[1][0] = WMMA_F32(a1, b0, acc[1][0]);
    acc[0][1] = WMMA_F32(a0, b1, acc[0][1]);
    acc[1][1] = WMMA_F32(a1, b1, acc[1][1]);
    acc[0][2] = WMMA_F32(a0, b2, acc[0][2]);
    acc[1][2] = WMMA_F32(a1, b2, acc[1][2]);
    acc[0][3] = WMMA_F32(a0, b3, acc[0][3]);
    acc[1][3] = WMMA_F32(a1, b3, acc[1][3]);

#pragma unroll
    for (int i = 0; i < 2; i++)
#pragma unroll
        for (int j = 0; j < 4; j++)
#pragma unroll
            for (int r = 0; r < 8; r++) {
                const int row = m0 + 16 * i + r + 8 * hlf;
                const int col = n0 + 16 * j + l16;
                float v = acc[i][j][r];
                if (bias) v += bias[col];
                Y[(size_t)row * Nd + col] = v;
            }
}

// ---------------------------------------------------------------------------
// RoPE in place on q and k sections of qkv[b][n][3*C]:
//   out[d]    = t[d]*cos[d]       - t[d+32]*sin[d]      (d < 32)
//   out[d+32] = t[d+32]*cos[d+32] + t[d]*sin[d+32]
// One thread per (b, n, h, i<32): total 2^22 threads.
// ---------------------------------------------------------------------------
__global__ __launch_bounds__(256)
void rope_kernel(float* __restrict__ qkv,
                 const float* __restrict__ cosT, const float* __restrict__ sinT)
{
    const int t = blockIdx.x * blockDim.x + threadIdx.x;
    const int i = t & 31;
    const int h = (t >> 5) & 15;
    const int n = (t >> 9) & (Nseq - 1);
    const int b = t >> 20;

    const size_t base = (size_t)(b * Nseq + n) * ThreeC + h * HD;
    const float c0 = cosT[n * HD + i],      s0 = sinT[n * HD + i];
    const float c1 = cosT[n * HD + i + 32], s1 = sinT[n * HD + i + 32];

    const float q0 = qkv[base + i], q1 = qkv[base + i + 32];
    qkv[base + i]      = q0 * c0 - q1 * s0;
    qkv[base + i + 32] = q1 * c1 + q0 * s1;

    const float k0 = qkv[base + Cdim + i], k1 = qkv[base + Cdim + i + 32];
    qkv[base + Cdim + i]      = k0 * c0 - k1 * s0;
    qkv[base + Cdim + i + 32] = k1 * c1 + k0 * s1;
}

// ---------------------------------------------------------------------------
// Flash attention, one wave per (b, h, 16-query tile).
// S^T = K_tile(16x64) x Q^T(64x16)  -> C layout: M=key (vgpr+half), N=query (lane)
// softmax stats live per-lane (query) -> pure scalar online update
// O^T = V^T(64x16keys) x P^T(16x16) -> C layout: M=headdim, N=query
// Pipelining: all 16 K A-frags batched before the S^T WMMA chain; all 32 V
// loads issued right after it so they overlap the softmax VALU work.
// ---------------------------------------------------------------------------
__global__ __launch_bounds__(256)
void flash_attn_kernel(const float* __restrict__ qkv, float* __restrict__ O)
{
    const int lane = threadIdx.x & 31;
    const int wid  = blockIdx.x * (blockDim.x >> 5) + (threadIdx.x >> 5);
    const int qt = wid & 127;          // N/16 = 128 query tiles
    const int h  = (wid >> 7) & 15;
    const int b  = wid >> 11;
    const int hlf = lane >> 4;
    const int l16 = lane & 15;
    const int q0  = qt * 16;
    const float scale = 0.125f;        // 1/sqrt(64)

    // Preload Q^T B-fragments: frag kk covers hd = 4kk..4kk+3
    const size_t rowQ = (size_t)(b * Nseq + q0 + l16) * ThreeC + h * HD;
    v2f qb[16];
#pragma unroll
    for (int kk = 0; kk < 16; kk++)
        qb[kk] = *(const v2f*)(qkv + rowQ + 4 * kk + 2 * hlf);

    v8f zero = {};
    v8f acc[4];
#pragma unroll
    for (int t = 0; t < 4; t++) acc[t] = zero;
    float m = -3.0e38f, l = 0.0f;

    const size_t baseK = (size_t)(b * Nseq) * ThreeC + Cdim + h * HD;
    const size_t baseV = (size_t)(b * Nseq) * ThreeC + 2 * Cdim + h * HD;

    for (int kt = 0; kt < Nseq / 16; kt++) {
        const int key0 = kt * 16;

        // Batch-load the 16 K A-fragments, then run the S^T WMMA chain.
        const size_t rowK = baseK + (size_t)(key0 + l16) * ThreeC;
        v2f ka[16];
#pragma unroll
        for (int kk = 0; kk < 16; kk++)
            ka[kk] = *(const v2f*)(qkv + rowK + 4 * kk + 2 * hlf);

        v8f s = zero;
#pragma unroll
        for (int kk = 0; kk < 16; kk++)
            s = WMMA_F32(ka[kk], qb[kk], s);

        // Issue all V loads now; they overlap the softmax VALU work below.
        v2f va[4][4];
#pragma unroll
        for (int t = 0; t < 4; t++) {
            const size_t colV = baseV + t * 16 + l16;
#pragma unroll
            for (int kk = 0; kk < 4; kk++) {
                const int ka0 = key0 + 4 * kk + 2 * hlf;
                va[t][kk].x = qkv[colV + (size_t)ka0 * ThreeC];
                va[t][kk].y = qkv[colV + (size_t)(ka0 + 1) * ThreeC];
            }
        }

        float sv[8];
#pragma unroll
        for (int r = 0; r < 8; r++) sv[r] = s[r] * scale;

        // per-query (lane) max over this tile's 16 keys
        float tm = sv[0];
#pragma unroll
        for (int r = 1; r < 8; r++) tm = fmaxf(tm, sv[r]);
        tm = fmaxf(tm, __shfl_xor(tm, 16, 32));

        const float mn   = fmaxf(m, tm);
        const float corr = __expf(m - mn);
        m = mn;

        float p[8];
        float rs = 0.0f;
#pragma unroll
        for (int r = 0; r < 8; r++) { p[r] = __expf(sv[r] - mn); rs += p[r]; }
        rs += __shfl_xor(rs, 16, 32);
        l = l * corr + rs;

#pragma unroll
        for (int t = 0; t < 4; t++)
#pragma unroll
            for (int r = 0; r < 8; r++) acc[t][r] *= corr;

        // Build P^T B-frags from S^T C-layout (8 half-swap shuffles)
        const float x2 = __shfl_xor(p[2], 16, 32), x3 = __shfl_xor(p[3], 16, 32);
        const float x6 = __shfl_xor(p[6], 16, 32), x7 = __shfl_xor(p[7], 16, 32);
        const float u0 = __shfl_xor(p[0], 16, 32), u1 = __shfl_xor(p[1], 16, 32);
        const float u4 = __shfl_xor(p[4], 16, 32), u5 = __shfl_xor(p[5], 16, 32);
        v2f pb[4];
        pb[0].x = hlf ? x2 : p[0];  pb[0].y = hlf ? x3 : p[1];
        pb[1].x = hlf ? x6 : p[4];  pb[1].y = hlf ? x7 : p[5];
        pb[2].x = hlf ? p[2] : u0;  pb[2].y = hlf ? p[3] : u1;
        pb[3].x = hlf ? p[6] : u4;  pb[3].y = hlf ? p[7] : u5;

        // O^T += V^T x P^T : 4 head-dim tiles x 4 K-steps (operands resident)
#pragma unroll
        for (int t = 0; t < 4; t++)
#pragma unroll
            for (int kk = 0; kk < 4; kk++)
                acc[t] = WMMA_F32(va[t][kk], pb[kk], acc[t]);
    }

    const float inv = 1.0f / l;
    const size_t baseO = (size_t)(b * Nseq + q0 + l16) * Cdim + h * HD;
#pragma unroll
    for (int t = 0; t < 4; t++)
#pragma unroll
        for (int r = 0; r < 8; r++)
            O[baseO + t * 16 + r + 8 * hlf] = acc[t][r] * inv;
}

// ---------------------------------------------------------------------------
extern "C" void kernel_launch(void* const* d_in, const int* in_sizes, int n_in,
                              void* d_out, int out_size, void* d_ws, size_t ws_size,
                              hipStream_t stream)
{
    const float* x        = (const float*)d_in[0];
    const float* rope_cos = (const float*)d_in[1];
    const float* rope_sin = (const float*)d_in[2];
    const float* W_qkv    = (const float*)d_in[3];
    const float* W_proj   = (const float*)d_in[4];
    const float* b_proj   = (const float*)d_in[5];
    float* out  = (float*)d_out;

    float* qkv  = (float*)d_ws;                           // 8192*3072 f32 = 96 MB
    float* attn = qkv + (size_t)Mrows * ThreeC;           // 8192*1024 f32 = 32 MB

    // 1) qkv = x @ W_qkv^T
    {
        const int waves = (Mrows / 32) * (ThreeC / 64);   // 12288
        gemm_xwt_kernel<<<waves / 8, 256, 0, stream>>>(x, W_qkv, nullptr, qkv,
                                                       Mrows, ThreeC, Cdim);
    }
    // 2) RoPE on q,k in place
    {
        const int threads = Bsz * Nseq * Hn * 32;         // 2^22
        rope_kernel<<<threads / 256, 256, 0, stream>>>(qkv, rope_cos, rope_sin);
    }
    // 3) flash attention -> attn
    {
        const int waves = Bsz * Hn * (Nseq / 16);         // 8192
        flash_attn_kernel<<<waves / 8, 256, 0, stream>>>(qkv, attn);
    }
    // 4) out = attn @ W_proj^T + b_proj
    {
        const int waves = (Mrows / 32) * (Cdim / 64);     // 4096
        gemm_xwt_kernel<<<waves / 8, 256, 0, stream>>>(attn, W_proj, b_proj, out,
                                                       Mrows, Cdim, Cdim);
    }
}